// OTFace_Arcface_20401094656245
// MI455X (gfx1250) — compile-verified
//
#include <hip/hip_runtime.h>
#include <hip/hip_bf16.h>
#include <math.h>

// ---------------- CDNA5 vector types for WMMA ----------------
typedef __attribute__((ext_vector_type(2))) float v2f;
typedef __attribute__((ext_vector_type(8))) float v8f;

// ---------------- problem constants ----------------
#define B_SZ 24
#define D_EMB 512
#define N_CLASSES 10000
#define NPTS 256
#define PDIM 196            // 14*14, = 49*4 exactly
#define CSTRIDE 257         // padded LDS row stride (bank-conflict free)

#define COS_M 0.9004471023526769f
#define SIN_M 0.43496553411123023f
#define TH_C (-0.9004471023526769f)
#define MM_C 0.19573449035005362f
#define SCALE_S 64.0f
#define LOG_MU (-5.545177444479562f)   // -log(256)

// =====================================================================
// 1) Row-normalize embeddings into padded 32x512 buffer (rows 24..31 = 0)
// =====================================================================
__global__ void k_norm_emb(const float* __restrict__ emb, float* __restrict__ embN) {
    int row = blockIdx.x;          // 0..31
    int t = threadIdx.x;           // 0..127
    __shared__ float red[128];
    float s = 0.f;
    if (row < B_SZ) {
        for (int k = t; k < D_EMB; k += 128) { float v = emb[row * D_EMB + k]; s += v * v; }
    }
    red[t] = s; __syncthreads();
    for (int off = 64; off > 0; off >>= 1) { if (t < off) red[t] += red[t + off]; __syncthreads(); }
    float inv = (row < B_SZ) ? (1.f / sqrtf(red[0])) : 0.f;
    for (int k = t; k < D_EMB; k += 128)
        embN[row * D_EMB + k] = (row < B_SZ) ? emb[row * D_EMB + k] * inv : 0.f;
}

// =====================================================================
// 2) Per-class inverse column norm of kernel (512 x 10000)
// =====================================================================
__global__ void k_col_invnorm(const float* __restrict__ kern, float* __restrict__ invn) {
    int n = blockIdx.x * 256 + threadIdx.x;
    if (n < N_CLASSES) {
        float s = 0.f;
        for (int k = 0; k < D_EMB; ++k) { float v = kern[(size_t)k * N_CLASSES + n]; s += v * v; }
        invn[n] = 1.f / sqrtf(s);
    }
}

// =====================================================================
// 3) ArcFace GEMM: cos = embN(24x512) @ kn(512x10000) via V_WMMA_F32_16X16X4
//    grid = 1250 waves (2 M-tiles x 625 N-tiles), block = 32 (one wave)
// =====================================================================
__global__ void k_arcface_wmma(const float* __restrict__ embN,
                               const float* __restrict__ kern,
                               const float* __restrict__ invn,
                               const int*  __restrict__ label,
                               float* __restrict__ outArc,
                               float* __restrict__ outOrig) {
    int tile  = blockIdx.x;        // 0..1249
    int mtile = tile & 1;
    int ntile = tile >> 1;         // 0..624 (625*16 == 10000 exactly)
    int lane  = threadIdx.x;       // 0..31
    int half  = lane >> 4;
    int l16   = lane & 15;

    int arow = mtile * 16 + l16;                  // A fragment row (padded to 32)
    int ncol = ntile * 16 + l16;                  // B fragment column
    float invc = invn[ncol];

    const float* aptr  = embN + (size_t)arow * D_EMB + 2 * half;
    const float* bbase = kern + ncol;             // column, stride N_CLASSES

    v8f acc = {};
#pragma unroll 4
    for (int k = 0; k < D_EMB; k += 4) {
        v2f a = *(const v2f*)(aptr + k);          // (M, k+2h), (M, k+2h+1)
        int kk = k + 2 * half;
        v2f b;
        b.x = bbase[(size_t)kk * N_CLASSES] * invc;
        b.y = bbase[(size_t)(kk + 1) * N_CLASSES] * invc;
        acc = __builtin_amdgcn_wmma_f32_16x16x4_f32(false, a, false, b,
                                                    (short)0, acc, false, false);
    }

    // C/D layout: VGPR r -> M = r + 8*half, N = lane&15
#pragma unroll
    for (int r = 0; r < 8; ++r) {
        int m = mtile * 16 + r + 8 * half;
        if (m < B_SZ) {
            int n = ntile * 16 + l16;
            float c = fminf(fmaxf(acc[r], -1.f), 1.f);
            outOrig[(size_t)m * N_CLASSES + n] = c * SCALE_S;
            float v = c;
            if (label[m] == n) {
                float sin_t = sqrtf(fmaxf(1.f - c * c, 0.f));
                float ctm = c * COS_M - sin_t * SIN_M;
                v = (c > TH_C) ? ctm : (c - MM_C);
            }
            outArc[(size_t)m * N_CLASSES + n] = v * SCALE_S;
        }
    }
}

// =====================================================================
// 4) dists = embN @ embN^T  (24x24) — one block of 576 threads
// =====================================================================
__global__ void k_emb_dists(const float* __restrict__ embN, float* __restrict__ dists) {
    int t = threadIdx.x;           // 0..575
    int i = t / B_SZ, j = t % B_SZ;
    float s = 0.f;
    for (int k = 0; k < D_EMB; ++k) s += embN[i * D_EMB + k] * embN[j * D_EMB + k];
    dists[t] = s;
}

// =====================================================================
// 5) Sinkhorn OT distance, one (a,b) pair per workgroup.
//    Cost matrix 256x257 f32 lives entirely in the 320KB WGP LDS.
//    Gram via V_WMMA_F32_16X16X4 (49 K-steps), then 6 log-domain iters.
//    Dynamic LDS = (256*257 + 5*256)*4 = 268288 bytes.
// =====================================================================
__global__ void k_sinkhorn_pair(const float* __restrict__ feats,   // 24 x 256 x 196
                                float* __restrict__ Dmat) {        // 576
    extern __shared__ float lds[];
    float* Cm  = lds;                       // 256*257
    float* x2s = lds + NPTS * CSTRIDE;      // 256
    float* y2s = x2s + NPTS;                // 256
    float* fv  = y2s + NPTS;                // 256
    float* gv  = fv  + NPTS;                // 256
    float* red = gv  + NPTS;                // 256

    int pair = blockIdx.x;
    int ai = pair / B_SZ, bi = pair % B_SZ;
    const float* xa = feats + (size_t)ai * NPTS * PDIM;
    const float* xb = feats + (size_t)bi * NPTS * PDIM;
    int t = threadIdx.x;                    // 0..255

    // ---- per-point squared norms; init potentials ----
    {
        const float* ra = xa + (size_t)t * PDIM;
        const float* rb = xb + (size_t)t * PDIM;
        float sa = 0.f, sb = 0.f;
        for (int d = 0; d < PDIM; ++d) { sa += ra[d] * ra[d]; sb += rb[d] * rb[d]; }
        x2s[t] = sa; y2s[t] = sb; fv[t] = 0.f; gv[t] = 0.f;
    }
    __syncthreads();

    // ---- Gram + cost matrix build: 16x16 tile grid, 8 waves ----
    int wave = t >> 5;
    int lane = t & 31;
    int half = lane >> 4;
    int l16  = lane & 15;
    for (int tile = wave; tile < 256; tile += 8) {
        int ti = tile >> 4, tj = tile & 15;
        const float* ap = xa + (size_t)(ti * 16 + l16) * PDIM + 2 * half;
        const float* bp = xb + (size_t)(tj * 16 + l16) * PDIM + 2 * half;
        v8f acc = {};
#pragma unroll 7
        for (int k = 0; k < PDIM; k += 4) {
            v2f a = *(const v2f*)(ap + k);
            v2f b = *(const v2f*)(bp + k);
            acc = __builtin_amdgcn_wmma_f32_16x16x4_f32(false, a, false, b,
                                                        (short)0, acc, false, false);
        }
#pragma unroll
        for (int r = 0; r < 8; ++r) {
            int m = ti * 16 + r + 8 * half;
            int n = tj * 16 + l16;
            float d2 = x2s[m] + y2s[n] - 2.f * acc[r];
            Cm[m * CSTRIDE + n] = sqrtf(fmaxf(d2, 0.f) + 1e-12f);
        }
    }
    __syncthreads();

    const float inv_eps = 10.f, eps = 0.1f;

    // ---- 6 log-domain Sinkhorn iterations, fully in LDS ----
    for (int it = 0; it < 6; ++it) {
        // f_i = -eps * (LSE_j((g_j - C_ij)/eps) + log_nu) ; thread t = row i
        {
            const float* crow = Cm + t * CSTRIDE;
            float mx = -3.0e38f;
            for (int j = 0; j < NPTS; ++j)
                mx = fmaxf(mx, (gv[j] - crow[j]) * inv_eps);
            float s = 0.f;
            for (int j = 0; j < NPTS; ++j)
                s += __expf((gv[j] - crow[j]) * inv_eps - mx);
            fv[t] = -eps * (mx + __logf(s) + LOG_MU);
        }
        __syncthreads();
        // g_j = -eps * (LSE_i((f_i - C_ij)/eps) + log_mu) ; thread t = col j
        {
            float mx = -3.0e38f;
            for (int i = 0; i < NPTS; ++i)
                mx = fmaxf(mx, (fv[i] - Cm[i * CSTRIDE + t]) * inv_eps);
            float s = 0.f;
            for (int i = 0; i < NPTS; ++i)
                s += __expf((fv[i] - Cm[i * CSTRIDE + t]) * inv_eps - mx);
            gv[t] = -eps * (mx + __logf(s) + LOG_MU);
        }
        __syncthreads();
    }

    // ---- <T, C> = sum exp((f+g-C)/eps + log_mu + log_nu) * C ----
    {
        const float* crow = Cm + t * CSTRIDE;
        float ft = fv[t];
        float local = 0.f;
        for (int j = 0; j < NPTS; ++j) {
            float c = crow[j];
            local += __expf((ft + gv[j] - c) * inv_eps + 2.f * LOG_MU) * c;
        }
        red[t] = local;
    }
    __syncthreads();
    for (int off = 128; off > 0; off >>= 1) {
        if (t < off) red[t] += red[t + off];
        __syncthreads();
    }
    if (t == 0) Dmat[pair] = red[0];
}

// =====================================================================
// 6) Triple statistics: n_cosine_pair, ot_pair, wass_loss
// =====================================================================
__global__ void k_finalize(const int* __restrict__ label,
                           const float* __restrict__ dists,
                           const float* __restrict__ Dmat,
                           float* __restrict__ outWass,
                           float* __restrict__ outNPair,
                           float* __restrict__ outOTPair) {
    __shared__ float rw[256];
    __shared__ int   rn[256];
    __shared__ int   ro[256];
    int t = threadIdx.x;
    int cn = 0, co = 0; float wass = 0.f;
    for (int idx = t; idx < B_SZ * B_SZ * B_SZ; idx += 256) {
        int a = idx / (B_SZ * B_SZ);
        int rem = idx % (B_SZ * B_SZ);
        int p = rem / B_SZ, n = rem % B_SZ;
        bool eq_ap = (label[a] == label[p]);
        bool eq_an = (label[a] == label[n]);
        float diff = dists[a * B_SZ + n] - dists[a * B_SZ + p];
        if (eq_ap && !eq_an && (diff > 0.f)) {
            cn++;
            float dd = Dmat[a * B_SZ + p] - Dmat[a * B_SZ + n];  // BETA=0
            if (dd > 0.f) { co++; wass += dd; }
        }
    }
    rw[t] = wass; rn[t] = cn; ro[t] = co;
    __syncthreads();
    for (int off = 128; off > 0; off >>= 1) {
        if (t < off) { rw[t] += rw[t + off]; rn[t] += rn[t + off]; ro[t] += ro[t + off]; }
        __syncthreads();
    }
    if (t == 0) {
        int ot = ro[0];
        float w = rw[0];
        *outWass   = (w > 0.f) ? (w / (float)(ot > 1 ? ot : 1)) : 0.f;
        *outNPair  = (float)rn[0];
        *outOTPair = (float)ot;
    }
}

// =====================================================================
// Launcher
// =====================================================================
extern "C" void kernel_launch(void* const* d_in, const int* in_sizes, int n_in,
                              void* d_out, int out_size, void* d_ws, size_t ws_size,
                              hipStream_t stream) {
    const float* emb   = (const float*)d_in[0];   // 24x512
    const float* feats = (const float*)d_in[1];   // 24x256x14x14
    const float* kern  = (const float*)d_in[2];   // 512x10000
    const int*   label = (const int*)  d_in[3];   // 24
    // d_in[4] = epoch (==1, always take the OT branch)

    float* out = (float*)d_out;
    float* outArc   = out;                                  // 240000
    float* outWass  = out + B_SZ * N_CLASSES;               // 1
    float* outOrig  = out + B_SZ * N_CLASSES + 1;           // 240000
    float* outNPair = out + 2 * B_SZ * N_CLASSES + 1;       // 1
    float* outOT    = out + 2 * B_SZ * N_CLASSES + 2;       // 1

    float* ws    = (float*)d_ws;
    float* embN  = ws;                     // 32*512 (padded, rows 24..31 zero)
    float* invn  = embN + 32 * D_EMB;      // 10000
    float* dists = invn + N_CLASSES;       // 576
    float* Dmat  = dists + B_SZ * B_SZ;    // 576

    k_norm_emb<<<32, 128, 0, stream>>>(emb, embN);
    k_col_invnorm<<<(N_CLASSES + 255) / 256, 256, 0, stream>>>(kern, invn);
    k_arcface_wmma<<<2 * (N_CLASSES / 16), 32, 0, stream>>>(embN, kern, invn, label,
                                                            outArc, outOrig);
    k_emb_dists<<<1, B_SZ * B_SZ, 0, stream>>>(embN, dists);

    // 268288 B dynamic LDS: cost matrix resident in the 320KB CDNA5 WGP LDS
    size_t ldsBytes = (size_t)(NPTS * CSTRIDE + 5 * NPTS) * sizeof(float);
    k_sinkhorn_pair<<<B_SZ * B_SZ, NPTS, ldsBytes, stream>>>(feats, Dmat);

    k_finalize<<<1, 256, 0, stream>>>(label, dists, Dmat, outWass, outNPair, outOT);
}